// InvariantPointAttention_3DROPE_74191265071611
// MI455X (gfx1250) — compile-verified
//
#include <hip/hip_runtime.h>
#include <hip/hip_bf16.h>
#include <math.h>

// ---------------- problem constants ----------------
#define NRES   1024
#define CSD    384
#define CZD    128
#define HD     8
#define GD     16
#define PQD    8
#define PVD    12
#define CD     48      // 3*G
#define CZ4D   32
#define INF_V  100000.0f
#define EPS_V  1e-8f
#define S3C    0.57735026918962576451f   // 1/sqrt(3)

typedef float v2f __attribute__((ext_vector_type(2)));
typedef float v4f __attribute__((ext_vector_type(4)));
typedef float v8f __attribute__((ext_vector_type(8)));
typedef int   v4i __attribute__((ext_vector_type(4)));

#if defined(__has_builtin)
#if __has_builtin(__builtin_amdgcn_global_load_async_to_lds_b128)
#define HAVE_ASYNC_LDS 1
#endif
#endif

// ---------------- WMMA f32 16x16x4 helpers ----------------
__device__ __forceinline__ v2f mk2(float x, float y) { v2f r; r.x = x; r.y = y; return r; }

__device__ __forceinline__ v8f wmma_f32_k4(v2f a, v2f b, v8f c) {
  // D = A(16x4) * B(4x16) + C(16x16), all f32, wave32
  return __builtin_amdgcn_wmma_f32_16x16x4_f32(false, a, false, b, (short)0, c, false, false);
}

// A fragment: row-major A, lda stride; tile origin (row0, k0).
// ISA layout: VGPR0 = K {0|2}, VGPR1 = K {1|3}; rows on lanes &15.
__device__ __forceinline__ v2f load_a_frag(const float* A, int lda, int row0, int k0, int lane) {
  int r  = row0 + (lane & 15);
  int kk = k0 + ((lane >> 4) << 1);
  const float* p = A + (size_t)r * lda + kk;
  return mk2(p[0], p[1]);
}

// B fragment: row-major B (K x N), ldb stride; tile origin (k0, col0).
__device__ __forceinline__ v2f load_b_frag(const float* B, int ldb, int k0, int col0, int lane) {
  int c  = col0 + (lane & 15);
  int kk = k0 + ((lane >> 4) << 1);
  return mk2(B[(size_t)kk * ldb + c], B[(size_t)(kk + 1) * ldb + c]);
}

// D tile: row r of acc vgpr -> matrix row r + 8*(lane>=16); col = lane&15.
__device__ __forceinline__ void store_d_tile(float* D, int ldd, int row0, int col0, int lane, v8f acc) {
  int col   = col0 + (lane & 15);
  int rbase = row0 + ((lane >> 4) << 3);
#pragma unroll
  for (int r = 0; r < 8; ++r) D[(size_t)(rbase + r) * ldd + col] = acc[r];
}

// ---------------- generic WMMA GEMM: D = A(MxK) @ B(KxN) + bias ----------------
// grid = (N/16, M/16), block = 32 (one wave per 16x16 tile)
__global__ void wmma_gemm_bias(const float* __restrict__ A, const float* __restrict__ B,
                               const float* __restrict__ bias, float* __restrict__ D,
                               int M, int K, int N) {
  int lane = threadIdx.x & 31;
  int col0 = blockIdx.x * 16;
  int row0 = blockIdx.y * 16;
  v8f acc = {};
  for (int k = 0; k < K; k += 4) {
    v2f a = load_a_frag(A, K, row0, k, lane);
    v2f b = load_b_frag(B, N, k, col0, lane);
    acc = wmma_f32_k4(a, b, acc);
  }
  float bv = bias[col0 + (lane & 15)];
#pragma unroll
  for (int r = 0; r < 8; ++r) acc[r] += bv;
  store_d_tile(D, N, row0, col0, lane, acc);
}

// ---------------- setup: fused z-projection weight/bias + hw scale ----------------
__global__ void setup_wz(const float* __restrict__ wb, const float* __restrict__ bb,
                         const float* __restrict__ wdz, const float* __restrict__ bdz,
                         const float* __restrict__ gw,
                         float* __restrict__ wz, float* __restrict__ wzb,
                         float* __restrict__ hw) {
  int tid = threadIdx.x;
  for (int i = tid; i < CZD * 48; i += blockDim.x) {
    int k = i / 48, c = i % 48;
    float v = 0.f;
    if (c < 8)       v = wb[k * HD + c];
    else if (c < 40) v = wdz[k * CZ4D + (c - 8)];
    wz[i] = v;
  }
  if (tid < 48) {
    float v = 0.f;
    if (tid < 8)       v = bb[tid];
    else if (tid < 40) v = bdz[tid - 8];
    wzb[tid] = v;
  }
  if (tid < GD) {
    // softplus(gw) * sqrt(1/G) * (1/sqrt(3)) folded into q scaling
    hw[tid] = log1pf(__expf(gw[tid])) * 0.25f * S3C;
  }
}

// ---------------- rotation / packing: qh, k_rot, v||v_pts ----------------
// grid = NRES, block = 128
__global__ void post_proj(const float* __restrict__ tmpQ, const float* __restrict__ tmpKV,
                          const float* __restrict__ tmpKVP, const float* __restrict__ rot,
                          const float* __restrict__ hw,
                          float* __restrict__ qbuf,   // (H, N, 48), hw/sqrt3 folded
                          float* __restrict__ kbuf,   // (H, N, 48)
                          float* __restrict__ vbuf) { // (H, N, 96): v(48) | v_pts(36) | pad(12)
  int n = blockIdx.x, tid = threadIdx.x;
  __shared__ float R[9];
  if (tid < 9) R[tid] = rot[n * 9 + tid];
  __syncthreads();

  for (int idx = tid; idx < HD * GD; idx += blockDim.x) {
    int h = idx >> 4, g = idx & 15;
    const float* q3 = tmpQ + (size_t)n * (HD * CD) + h * CD + g * 3;
    const float* k3 = tmpKV + (size_t)n * (2 * HD * CD) + h * CD + g * 3;
    float sc = hw[g];
    size_t qo = ((size_t)h * NRES + n) * CD + g * 3;
#pragma unroll
    for (int i = 0; i < 3; ++i) {
      float qi = R[i * 3 + 0] * q3[0] + R[i * 3 + 1] * q3[1] + R[i * 3 + 2] * q3[2];
      float ki = R[i * 3 + 0] * k3[0] + R[i * 3 + 1] * k3[1] + R[i * 3 + 2] * k3[2];
      qbuf[qo + i] = qi * sc;
      kbuf[qo + i] = ki;
    }
  }
  // v part
  for (int c = tid; c < HD * CD; c += blockDim.x) {
    int h = c / CD, cc = c % CD;
    vbuf[((size_t)h * NRES + n) * 96 + cc] =
        tmpKV[(size_t)n * (2 * HD * CD) + HD * CD + h * CD + cc];
  }
  // v_pts part + zero pad
  for (int i = tid; i < HD * 48; i += blockDim.x) {
    int h = i / 48, c2 = i % 48;
    float val = 0.f;
    if (c2 < 36) {
      int comp = c2 / 12, p = c2 % 12;
      val = tmpKVP[(size_t)n * 480 + comp * 160 + h * 20 + PQD + p];
    }
    vbuf[((size_t)h * NRES + n) * 96 + 48 + c2] = val;
  }
}

// ---------------- fused z pass: logits-bias + pair_z, single read of z ----------------
// Z viewed as (N*N x 128); W = [wb | wdz | 0] (128 x 48).
// grid = (N*N/16)/8 blocks, block = 256 (8 waves, one 16-row tile each).
// z tile staged to LDS via async-load-to-LDS (CDNA5) when available.
#define ZT_ST 132   // padded LDS row stride (floats): (132 % 64) = 4 -> conflict-free A frags
__global__ void zproj(const float* __restrict__ z, const float* __restrict__ wz,
                      const float* __restrict__ wzb, const float* __restrict__ mask,
                      float* __restrict__ abuf,    // (H, N, N): sqrt(1/3)*b + mask term
                      float* __restrict__ pzbuf) { // (N, N, 32)
  __shared__ float zt[8][16 * ZT_ST];   // per-wave z tile (8448 B each)
  __shared__ float wzl[CZD * 48];       // fused weight, LDS-resident
  __shared__ float wzbl[48];
  int lane = threadIdx.x & 31;
  int wv   = threadIdx.x >> 5;

  for (int i = threadIdx.x; i < CZD * 48; i += 256) wzl[i] = wz[i];
  if (threadIdx.x < 48) wzbl[threadIdx.x] = wzb[threadIdx.x];
  __syncthreads();

  int rt = blockIdx.x * 8 + wv;             // 16-row tile index over N*N rows
  int n  = rt >> 6;                         // 1024/16 = 64 tiles per n
  int m0 = (rt & 63) << 4;
  const float* A = z + ((size_t)n * NRES + m0) * CZD;  // 2048 contiguous floats

  // prefetch next block's tile region for this wave (speculative, streaming)
  __builtin_prefetch(A + 8 * 16 * CZD, 0, 0);

  float* lt = zt[wv];
#pragma unroll
  for (int c = 0; c < 4; ++c) {
    int fi  = (c * 32 + lane) * 4;          // flat float index within tile
    int row = fi >> 7, col = fi & 127;
#if defined(HAVE_ASYNC_LDS)
    __builtin_amdgcn_global_load_async_to_lds_b128(
        (__attribute__((address_space(1))) v4i*)(A + fi),
        (__attribute__((address_space(3))) v4i*)&lt[row * ZT_ST + col], 0, 0);
#else
    *(v4f*)&lt[row * ZT_ST + col] = *(const v4f*)(A + fi);
#endif
  }
#if defined(HAVE_ASYNC_LDS)
  asm volatile("s_wait_asynccnt 0x0" ::: "memory");
#endif

  v8f acc0 = {}, acc1 = {}, acc2 = {};
  for (int k = 0; k < CZD; k += 4) {
    int kk = k + ((lane >> 4) << 1);
    v2f a  = mk2(lt[(lane & 15) * ZT_ST + kk], lt[(lane & 15) * ZT_ST + kk + 1]);
    int cb = lane & 15;
    v2f b0 = mk2(wzl[kk * 48 + cb],      wzl[(kk + 1) * 48 + cb]);
    v2f b1 = mk2(wzl[kk * 48 + 16 + cb], wzl[(kk + 1) * 48 + 16 + cb]);
    v2f b2 = mk2(wzl[kk * 48 + 32 + cb], wzl[(kk + 1) * 48 + 32 + cb]);
    acc0 = wmma_f32_k4(a, b0, acc0);
    acc1 = wmma_f32_k4(a, b1, acc1);
    acc2 = wmma_f32_k4(a, b2, acc2);
  }
  float mn = mask[n];
  int rbase = (lane >> 4) << 3;
#pragma unroll
  for (int t = 0; t < 3; ++t) {
    v8f acc = (t == 0) ? acc0 : (t == 1) ? acc1 : acc2;
    int col = t * 16 + (lane & 15);
    float bias = wzbl[col];
#pragma unroll
    for (int r = 0; r < 8; ++r) {
      int m = m0 + rbase + r;
      float val = acc[r] + bias;
      if (col < 8) {
        float mt = INF_V * (mn * mask[m] - 1.0f);
        abuf[((size_t)col * NRES + n) * NRES + m] = S3C * val + mt;
      } else if (col < 40) {
        pzbuf[((size_t)n * NRES + m) * CZ4D + (col - 8)] = val;
      }
    }
  }
}

// ---------------- fused attention: logits += qk, softmax, o/o_pt, probs -> abuf ----------------
// grid = (64 nblocks, 8 heads), block = 256 (8 waves). Logit tile lives in LDS.
#define LA_ST 1028  // padded stride: (1028 % 64) = 4 -> conflict-free A frags
__global__ void attn_fused(const float* __restrict__ qbuf, const float* __restrict__ kbuf,
                           const float* __restrict__ vbuf,
                           float* __restrict__ abuf, float* __restrict__ feats) {
  __shared__ float la[16 * LA_ST];
  __shared__ float lq[16 * CD];
  __shared__ float red0[256];
  __shared__ float red1[32];
  int lane = threadIdx.x & 31;
  int wv   = threadIdx.x >> 5;
  int n0   = blockIdx.x * 16;
  int h    = blockIdx.y;

  const float* qsrc = qbuf + ((size_t)h * NRES + n0) * CD;
  for (int i = threadIdx.x; i < 16 * CD; i += 256) lq[i] = qsrc[i];
  __syncthreads();

  // ---- logits: qk WMMA + (sqrt(1/3) b + mask) term from abuf ----
  const float* kh   = kbuf + (size_t)h * NRES * CD;
  const float* brow = abuf + (size_t)h * NRES * NRES + (size_t)n0 * NRES;
  for (int mt = wv; mt < 64; mt += 8) {
    int m0 = mt * 16;
    v8f acc = {};
    for (int k = 0; k < CD; k += 4) {
      int kk = k + ((lane >> 4) << 1);
      v2f a = mk2(lq[(lane & 15) * CD + kk], lq[(lane & 15) * CD + kk + 1]);
      const float* kp = kh + (size_t)(m0 + (lane & 15)) * CD + kk;
      v2f b = mk2(kp[0], kp[1]);
      acc = wmma_f32_k4(a, b, acc);
    }
    int col   = m0 + (lane & 15);
    int rbase = (lane >> 4) << 3;
#pragma unroll
    for (int r = 0; r < 8; ++r) {
      int row = rbase + r;
      la[row * LA_ST + col] = acc[r] + brow[(size_t)row * NRES + col];
    }
  }
  __syncthreads();

  // ---- softmax over the 16 rows (16 threads per row) ----
  {
    int row = threadIdx.x >> 4, t = threadIdx.x & 15;
    float* lr = la + row * LA_ST;
    float mx = -INFINITY;
    for (int i = t; i < NRES; i += 16) mx = fmaxf(mx, lr[i]);
    red0[threadIdx.x] = mx;
    __syncthreads();
    if (t == 0) {
      float m2 = red0[row * 16];
      for (int j = 1; j < 16; ++j) m2 = fmaxf(m2, red0[row * 16 + j]);
      red1[row] = m2;
    }
    __syncthreads();
    mx = red1[row];
    float sum = 0.f;
    for (int i = t; i < NRES; i += 16) { float e = __expf(lr[i] - mx); lr[i] = e; sum += e; }
    red0[threadIdx.x] = sum;
    __syncthreads();
    if (t == 0) {
      float s2 = 0.f;
      for (int j = 0; j < 16; ++j) s2 += red0[row * 16 + j];
      red1[16 + row] = 1.0f / s2;
    }
    __syncthreads();
    float inv = red1[16 + row];
    for (int i = t; i < NRES; i += 16) lr[i] *= inv;
  }
  __syncthreads();

  // ---- write probabilities back to abuf (consumed by o_pair) ----
  {
    float* adst = abuf + (size_t)h * NRES * NRES + (size_t)n0 * NRES;
    for (int i = threadIdx.x; i < 16 * NRES; i += 256) {
      int r = i >> 10, c = i & 1023;
      adst[i] = la[r * LA_ST + c];
    }
  }

  // ---- o / o_pt: probs(16x1024) @ [v|v_pts](1024x96), waves 0..5 ----
  if (wv < 6) {
    int col0 = wv * 16;
    const float* vh = vbuf + (size_t)h * NRES * 96;
    v8f acc = {};
    for (int k = 0; k < NRES; k += 4) {
      int kk = k + ((lane >> 4) << 1);
      v2f a = mk2(la[(lane & 15) * LA_ST + kk], la[(lane & 15) * LA_ST + kk + 1]);
      int c = col0 + (lane & 15);
      v2f b = mk2(vh[(size_t)kk * 96 + c], vh[(size_t)(kk + 1) * 96 + c]);
      acc = wmma_f32_k4(a, b, acc);
    }
    int col   = col0 + (lane & 15);
    int rbase = (lane >> 4) << 3;
    int fcol  = -1;
    if (col < 48) fcol = h * 48 + col;                 // o
    else {
      int c2 = col - 48;
      if (c2 < 36) { int comp = c2 / 12, p = c2 % 12; fcol = 384 + comp * 96 + h * 12 + p; } // o_pt
    }
    if (fcol >= 0) {
#pragma unroll
      for (int r = 0; r < 8; ++r) feats[(size_t)(n0 + rbase + r) * 1024 + fcol] = acc[r];
    }
  }
}

// ---------------- o_pair: per residue n, probs(8x1024) @ pair_z[n](1024x32) ----------------
// grid = N, block = 64 (2 waves, one 16-col tile each)
__global__ void out_opair(const float* __restrict__ abuf, const float* __restrict__ pzbuf,
                          float* __restrict__ feats) {
  int lane = threadIdx.x & 31;
  int ct   = threadIdx.x >> 5;
  int n    = blockIdx.x;
  const float* pz = pzbuf + (size_t)n * NRES * CZ4D;

  v8f acc = {};
  int row = lane & 15;   // head (rows 8..15 padded with zeros)
  for (int k = 0; k < NRES; k += 4) {
    int kk = k + ((lane >> 4) << 1);
    float ax = 0.f, ay = 0.f;
    if (row < 8) {
      const float* ap = abuf + (size_t)row * NRES * NRES + (size_t)n * NRES + kk;
      ax = ap[0]; ay = ap[1];
    }
    v2f a = mk2(ax, ay);
    int c = ct * 16 + (lane & 15);
    v2f b = mk2(pz[(size_t)kk * CZ4D + c], pz[(size_t)(kk + 1) * CZ4D + c]);
    acc = wmma_f32_k4(a, b, acc);
  }
  if (lane < 16) { // rows 0..7 hold the 8 heads
    int c = ct * 16 + lane;
#pragma unroll
    for (int r = 0; r < 8; ++r) feats[(size_t)n * 1024 + 768 + r * CZ4D + c] = acc[r];
  }
}

// ---------------- point norms ----------------
// grid = N, block = 96
__global__ void point_norms(float* __restrict__ feats) {
  int n = blockIdx.x, t = threadIdx.x; // t = h*12+p
  float* f = feats + (size_t)n * 1024;
  float x = f[384 + t], y = f[480 + t], zc = f[576 + t];
  f[672 + t] = sqrtf(x * x + y * y + zc * zc + EPS_V);
}

// ---------------- launcher ----------------
extern "C" void kernel_launch(void* const* d_in, const int* in_sizes, int n_in,
                              void* d_out, int out_size, void* d_ws, size_t ws_size,
                              hipStream_t stream) {
  (void)in_sizes; (void)n_in; (void)out_size; (void)ws_size;
  const float* s    = (const float*)d_in[0];
  const float* z    = (const float*)d_in[1];
  const float* rot  = (const float*)d_in[2];
  const float* mask = (const float*)d_in[3];
  const float* wq   = (const float*)d_in[4];
  const float* bq   = (const float*)d_in[5];
  const float* wkv  = (const float*)d_in[6];
  const float* bkv  = (const float*)d_in[7];
  const float* wkvp = (const float*)d_in[8];
  const float* bkvp = (const float*)d_in[9];
  const float* wb   = (const float*)d_in[10];
  const float* bb   = (const float*)d_in[11];
  const float* wdz  = (const float*)d_in[12];
  const float* bdz  = (const float*)d_in[13];
  const float* wout = (const float*)d_in[14];
  const float* bout = (const float*)d_in[15];
  const float* gw   = (const float*)d_in[16];
  float* out = (float*)d_out;

  float* ws    = (float*)d_ws;
  float* wz    = ws;  ws += 128 * 48;
  float* wzb   = ws;  ws += 64;
  float* hw    = ws;  ws += 64;
  float* tmpQ  = ws;  ws += (size_t)NRES * 384;
  float* tmpKV = ws;  ws += (size_t)NRES * 768;
  float* tmpKVP= ws;  ws += (size_t)NRES * 480;
  float* qbuf  = ws;  ws += (size_t)HD * NRES * CD;
  float* kbuf  = ws;  ws += (size_t)HD * NRES * CD;
  float* vbuf  = ws;  ws += (size_t)HD * NRES * 96;
  float* abuf  = ws;  ws += (size_t)HD * NRES * NRES;           // 32 MB
  float* pzbuf = ws;  ws += (size_t)NRES * NRES * CZ4D;         // 128 MB
  float* feats = ws;  ws += (size_t)NRES * 1024;                // 4 MB

  // 1. setup fused z-weight / hw
  setup_wz<<<1, 256, 0, stream>>>(wb, bb, wdz, bdz, gw, wz, wzb, hw);

  // 2-4. projections (WMMA f32)
  wmma_gemm_bias<<<dim3(384 / 16, NRES / 16), 32, 0, stream>>>(s, wq,   bq,   tmpQ,   NRES, CSD, 384);
  wmma_gemm_bias<<<dim3(768 / 16, NRES / 16), 32, 0, stream>>>(s, wkv,  bkv,  tmpKV,  NRES, CSD, 768);
  wmma_gemm_bias<<<dim3(480 / 16, NRES / 16), 32, 0, stream>>>(s, wkvp, bkvp, tmpKVP, NRES, CSD, 480);

  // 5. rotate q/k, pack v||v_pts
  post_proj<<<NRES, 128, 0, stream>>>(tmpQ, tmpKV, tmpKVP, rot, hw, qbuf, kbuf, vbuf);

  // 6. single pass over z (512 MB): bias+mask term -> abuf, pair_z -> pzbuf
  zproj<<<(NRES * NRES / 16) / 8, 256, 0, stream>>>(z, wz, wzb, mask, abuf, pzbuf);

  // 7. fused: logits += qk, softmax, o/o_pt, probs back to abuf
  attn_fused<<<dim3(NRES / 16, HD), 256, 0, stream>>>(qbuf, kbuf, vbuf, abuf, feats);

  // 8. o_pair
  out_opair<<<NRES, 64, 0, stream>>>(abuf, pzbuf, feats);

  // 9. norms
  point_norms<<<NRES, 96, 0, stream>>>(feats);

  // 10. final projection: feats(1024x1024) @ wout(1024x384) + bout
  wmma_gemm_bias<<<dim3(384 / 16, NRES / 16), 32, 0, stream>>>(feats, wout, bout, out, NRES, 1024, 384);
}